// HypergraphGAT_9225589752583
// MI455X (gfx1250) — compile-verified
//
#include <hip/hip_runtime.h>
#include <hip/hip_bf16.h>

// ---------------- Problem constants (match reference) ----------------
#define NNODES   50000
#define NEDGES   800000
#define ETOT     (NEDGES + NNODES)   // edges + self loops
#define INDIM    128
#define HIDC     32
#define NHEADS   4
#define OUTDIM   64
#define NGRAPHS  8
#define NEGSLOPE 0.2f
#define EPSA     1e-16f

typedef __attribute__((ext_vector_type(2))) float v2f;
typedef __attribute__((ext_vector_type(8))) float v8f;

// Orderable float <-> uint mapping for atomic float max
__device__ __forceinline__ unsigned f2ord(float f) {
    unsigned u = __float_as_uint(f);
    return (u & 0x80000000u) ? ~u : (u | 0x80000000u);
}
__device__ __forceinline__ float ord2f(unsigned u) {
    u = (u & 0x80000000u) ? (u & 0x7FFFFFFFu) : ~u;
    return __uint_as_float(u);
}

// ---------------------------------------------------------------------
// H[N,OUTC] = X[N,K] @ W[K,OUTC] using V_WMMA_F32_16X16X4_F32.
// One wave32 per 16x16 output tile; K stepped by 4 (32 WMMAs for K=128).
// A 16x4 f32 layout: lane m=lane&15 holds K = 2*(lane>>4)+{0,1} (2 VGPRs)
// B 4x16 f32 layout: lane n=lane&15 holds col n, same K split (2 VGPRs)
// C/D 16x16 f32:     VGPR r -> row 8*(lane>>4)+r, col = lane&15
// N, K, OUTC all multiples of 16 here (50000 = 3125*16).
// ---------------------------------------------------------------------
__global__ void gat_gemm_wmma(const float* __restrict__ X,
                              const float* __restrict__ W,
                              float* __restrict__ H,
                              int N, int K, int OUTC) {
    const int wavesPerBlock = blockDim.x >> 5;
    const int waveId = blockIdx.x * wavesPerBlock + (threadIdx.x >> 5);
    const int lane = threadIdx.x & 31;
    const int colTiles = OUTC >> 4;
    const int rowTile = waveId / colTiles;
    const int colTile = waveId - rowTile * colTiles;
    if (rowTile * 16 >= N) return;              // whole-wave uniform exit

    const int m    = rowTile * 16 + (lane & 15);
    const int n    = colTile * 16 + (lane & 15);
    const int koff = (lane >> 4) << 1;          // 0 or 2

    const float* __restrict__ xrow = X + (size_t)m * K;

    v8f acc = {};
    for (int kb = 0; kb < K; kb += 4) {
        v2f a, b;
        a.x = xrow[kb + koff];
        a.y = xrow[kb + koff + 1];
        b.x = W[(size_t)(kb + koff) * OUTC + n];
        b.y = W[(size_t)(kb + koff + 1) * OUTC + n];
        // 8 args: (neg_a, A, neg_b, B, c_mod, C, reuse_a, reuse_b)
        acc = __builtin_amdgcn_wmma_f32_16x16x4_f32(
            false, a, false, b, (short)0, acc, false, false);
    }

    const int rowBase = rowTile * 16 + ((lane >> 4) << 3);
    const int col     = colTile * 16 + (lane & 15);
#pragma unroll
    for (int r = 0; r < 8; ++r) {
        H[(size_t)(rowBase + r) * OUTC + col] = acc[r];
    }
}

// alpha_src[n,h] = <h[n,h,:], a_s[h,:]>, alpha_dst likewise
__global__ void attn_coef(const float* __restrict__ Hm,
                          const float* __restrict__ a_s,
                          const float* __restrict__ a_d,
                          float* __restrict__ as_out,
                          float* __restrict__ ad_out,
                          int N, int heads, int C) {
    int i = blockIdx.x * blockDim.x + threadIdx.x;
    if (i >= N * heads) return;
    int n = i / heads, h = i - n * heads;
    const float* row = Hm + (size_t)n * heads * C + (size_t)h * C;
    const float* vs  = a_s + (size_t)h * C;
    const float* vd  = a_d + (size_t)h * C;
    float ss = 0.f, sd = 0.f;
    for (int c = 0; c < C; ++c) {
        float v = row[c];
        ss = fmaf(v, vs[c], ss);
        sd = fmaf(v, vd[c], sd);
    }
    as_out[i] = ss;
    ad_out[i] = sd;
}

__device__ __forceinline__ void edge_sd(const int* __restrict__ src,
                                        const int* __restrict__ dst,
                                        int e, int& s, int& d) {
    if (e < NEDGES) { s = src[e]; d = dst[e]; }
    else            { s = e - NEDGES; d = s; }   // self loop
}

// Pass 1: segment max of leaky_relu(as[src]+ad[dst]) over dst
__global__ void edge_max(const int* __restrict__ src, const int* __restrict__ dst,
                         const float* __restrict__ as_, const float* __restrict__ ad_,
                         unsigned* __restrict__ nmax, int heads) {
    int i = blockIdx.x * blockDim.x + threadIdx.x;
    if (i >= ETOT * heads) return;
    int e = i / heads, h = i - e * heads;
    int s, d; edge_sd(src, dst, e, s, d);
    float el = as_[s * heads + h] + ad_[d * heads + h];
    el = (el >= 0.f) ? el : NEGSLOPE * el;
    atomicMax(&nmax[d * heads + h], f2ord(el));
}

// Pass 2: e_exp = exp(e - max[dst]); segment sum over dst
__global__ void edge_exp_sum(const int* __restrict__ src, const int* __restrict__ dst,
                             const float* __restrict__ as_, const float* __restrict__ ad_,
                             const unsigned* __restrict__ nmax,
                             float* __restrict__ eexp, float* __restrict__ nsum,
                             int heads) {
    int i = blockIdx.x * blockDim.x + threadIdx.x;
    if (i >= ETOT * heads) return;
    int e = i / heads, h = i - e * heads;
    int s, d; edge_sd(src, dst, e, s, d);
    float el = as_[s * heads + h] + ad_[d * heads + h];
    el = (el >= 0.f) ? el : NEGSLOPE * el;
    float ex = __expf(el - ord2f(nmax[d * heads + h]));
    eexp[i] = ex;
    atomicAdd(&nsum[d * heads + h], ex);
}

// Pass 3: out[dst,h,c] += h[src,h,c] * (e_exp / (sum[dst,h]+eps))
__global__ void edge_agg(const int* __restrict__ src, const int* __restrict__ dst,
                         const float* __restrict__ Hm,
                         const float* __restrict__ eexp,
                         const float* __restrict__ nsum,
                         float* __restrict__ out, int heads, int C) {
    long long total = (long long)ETOT * heads * C;
    long long stride = (long long)gridDim.x * blockDim.x;
    for (long long i = (long long)blockIdx.x * blockDim.x + threadIdx.x;
         i < total; i += stride) {
        int c = (int)(i % C);
        long long t = i / C;
        int h = (int)(t % heads);
        int e = (int)(t / heads);
        int s, d; edge_sd(src, dst, e, s, d);
        float alpha = eexp[(long long)e * heads + h] /
                      (nsum[d * heads + h] + EPSA);
        float v = Hm[(size_t)s * heads * C + (size_t)h * C + c] * alpha;
        atomicAdd(&out[(size_t)d * heads * C + (size_t)h * C + c], v);
    }
}

// out += bias (broadcast over nodes); optional ELU (alpha=1)
__global__ void bias_act(float* __restrict__ out, const float* __restrict__ bias,
                         int N, int F, int do_elu) {
    int i = blockIdx.x * blockDim.x + threadIdx.x;
    if (i >= N * F) return;
    float v = out[i] + bias[i % F];
    if (do_elu) v = (v > 0.f) ? v : (__expf(v) - 1.f);
    out[i] = v;
}

// Global mean pool: accumulate per-graph sums and counts
__global__ void pool_accum(const float* __restrict__ H3, const int* __restrict__ batch,
                           float* __restrict__ psum, float* __restrict__ pcnt,
                           int N, int F) {
    int i = blockIdx.x * blockDim.x + threadIdx.x;
    if (i >= N * F) return;
    int n = i / F, j = i - n * F;
    int g = batch[n];
    atomicAdd(&psum[g * F + j], H3[i]);
    if (j == 0) atomicAdd(&pcnt[g], 1.0f);
}

__global__ void pool_div(const float* __restrict__ psum, const float* __restrict__ pcnt,
                         float* __restrict__ out, int G, int F) {
    int i = blockIdx.x * blockDim.x + threadIdx.x;
    if (i >= G * F) return;
    float c = pcnt[i / F];
    out[i] = psum[i] / fmaxf(c, 1.0f);
}

// ---------------------------------------------------------------------
static inline int cdiv(long long a, long long b) { return (int)((a + b - 1) / b); }

extern "C" void kernel_launch(void* const* d_in, const int* in_sizes, int n_in,
                              void* d_out, int out_size, void* d_ws, size_t ws_size,
                              hipStream_t stream) {
    (void)in_sizes; (void)n_in; (void)out_size; (void)ws_size;

    const float* x    = (const float*)d_in[0];
    const int*   eidx = (const int*)d_in[1];           // [2, E]
    const int*   batch = (const int*)d_in[2];
    const float* W1 = (const float*)d_in[3];
    const float* a1s = (const float*)d_in[4];
    const float* a1d = (const float*)d_in[5];
    const float* b1 = (const float*)d_in[6];
    const float* W2 = (const float*)d_in[7];
    const float* a2s = (const float*)d_in[8];
    const float* a2d = (const float*)d_in[9];
    const float* b2 = (const float*)d_in[10];
    const float* W3 = (const float*)d_in[11];
    const float* a3s = (const float*)d_in[12];
    const float* a3d = (const float*)d_in[13];
    const float* b3 = (const float*)d_in[14];
    float* out = (float*)d_out;

    const int* src = eidx;
    const int* dst = eidx + NEDGES;

    // ---- workspace carve ----
    size_t off = 0;
    char* base = (char*)d_ws;
    auto carve = [&](size_t bytes) -> void* {
        void* p = base + off;
        off += (bytes + 255) & ~(size_t)255;
        return p;
    };
    float*    bufA = (float*)carve((size_t)NNODES * 128 * 4);
    float*    bufB = (float*)carve((size_t)NNODES * 128 * 4);
    float*    asA  = (float*)carve((size_t)NNODES * NHEADS * 4);
    float*    adA  = (float*)carve((size_t)NNODES * NHEADS * 4);
    unsigned* nmax = (unsigned*)carve((size_t)NNODES * NHEADS * 4);
    float*    nsum = (float*)carve((size_t)NNODES * NHEADS * 4);
    float*    eexp = (float*)carve((size_t)ETOT * NHEADS * 4);
    float*    psum = (float*)carve((size_t)NGRAPHS * OUTDIM * 4);
    float*    pcnt = (float*)carve((size_t)NGRAPHS * 4);

    const int TB = 256;

    auto run_layer = [&](const float* X, const float* W, const float* a_s,
                         const float* a_d, const float* bias,
                         float* Hproj, float* Agg,
                         int K, int heads, int C, int do_elu) {
        const int F = heads * C;
        // zero aggregation output
        hipMemsetAsync(Agg, 0, (size_t)NNODES * F * 4, stream);
        // projection GEMM (WMMA f32 16x16x4)
        {
            long long tiles = (long long)(NNODES / 16) * (F / 16);
            int blocks = cdiv(tiles, TB / 32);   // 8 waves per block
            gat_gemm_wmma<<<blocks, TB, 0, stream>>>(X, W, Hproj, NNODES, K, F);
        }
        // attention coefficients
        attn_coef<<<cdiv((long long)NNODES * heads, TB), TB, 0, stream>>>(
            Hproj, a_s, a_d, asA, adA, NNODES, heads, C);
        // segment softmax over incoming edges of each dst
        hipMemsetAsync(nmax, 0, (size_t)NNODES * heads * 4, stream); // ord(-inf) < any
        hipMemsetAsync(nsum, 0, (size_t)NNODES * heads * 4, stream);
        edge_max<<<cdiv((long long)ETOT * heads, TB), TB, 0, stream>>>(
            src, dst, asA, adA, nmax, heads);
        edge_exp_sum<<<cdiv((long long)ETOT * heads, TB), TB, 0, stream>>>(
            src, dst, asA, adA, nmax, eexp, nsum, heads);
        // weighted scatter aggregation
        edge_agg<<<8192, TB, 0, stream>>>(src, dst, Hproj, eexp, nsum, Agg, heads, C);
        // bias (+ optional ELU)
        bias_act<<<cdiv((long long)NNODES * F, TB), TB, 0, stream>>>(
            Agg, bias, NNODES, F, do_elu);
    };

    // Layer 1: x[.,128] -> bufB[.,128] (HEADS=4, C=32, concat, ELU)
    run_layer(x,    W1, a1s, a1d, b1, bufA, bufB, INDIM,       NHEADS, HIDC,  1);
    // Layer 2: bufB -> bufA proj, agg into bufA? no: proj bufA, agg bufB is input.
    // proj into bufA, aggregate into... bufB is still the input until attn done;
    // after gemm+attn we only read bufA, so aggregate over bufA into bufB is unsafe
    // (bufB is X for gemm only -> gemm completes before edge_agg on the stream).
    run_layer(bufB, W2, a2s, a2d, b2, bufA, bufB, NHEADS*HIDC, NHEADS, HIDC,  1);
    // Layer 3: bufB[.,128] -> proj bufA[.,64], agg bufB[.,64] (heads=1, no ELU)
    run_layer(bufB, W3, a3s, a3d, b3, bufA, bufB, NHEADS*HIDC, 1,      OUTDIM, 0);

    // Global mean pool -> out[8,64]
    hipMemsetAsync(psum, 0, (size_t)NGRAPHS * OUTDIM * 4, stream);
    hipMemsetAsync(pcnt, 0, (size_t)NGRAPHS * 4, stream);
    pool_accum<<<cdiv((long long)NNODES * OUTDIM, TB), TB, 0, stream>>>(
        bufB, batch, psum, pcnt, NNODES, OUTDIM);
    pool_div<<<cdiv((long long)NGRAPHS * OUTDIM, TB), TB, 0, stream>>>(
        psum, pcnt, out, NGRAPHS, OUTDIM);
}